// LinearAttention_17935783428544
// MI455X (gfx1250) — compile-verified
//
#include <hip/hip_runtime.h>
#include <math.h>

#define HEADS 4
#define DIM_HEAD 64
#define NB 16        // batch
#define CCH 256      // channels
#define NPIX 4096    // 64*64
#define O3 768       // 3 * HEADS * DIM_HEAD
#define QSCALE 0.125f // 64^-0.5
#define PAD 68       // LDS row pitch (floats): 16B-aligned rows, conflict-free stride

typedef __attribute__((ext_vector_type(2))) float v2f;
typedef __attribute__((ext_vector_type(8))) float v8f;

// D = A(16x4,f32) * B(4x16,f32) + C(16x16,f32)  -> v_wmma_f32_16x16x4_f32
__device__ __forceinline__ v8f wmma4(v2f a, v2f b, v8f c) {
    return __builtin_amdgcn_wmma_f32_16x16x4_f32(
        /*neg_a=*/false, a, /*neg_b=*/false, b,
        /*c_mod=*/(short)0, c, /*reuse_a=*/false, /*reuse_b=*/false);
}

__device__ __forceinline__ unsigned lds_off(const void* p) {
    // generic LDS address carries the LDS byte offset in addr[31:0] (ISA 10.2)
    return (unsigned)(unsigned long long)p;
}

// ---------------------------------------------------------------------------
// Kernel 1: qkv[b,o,n] = sum_c w_qkv[o,c] * x[b,c,n]
// one wave -> 32(o) x 64(n) tile: 8 WMMA per k-step from 2 b64 + 8 b32 loads.
// waves = 16 * 24 * 64 = 24576; 3072 blocks
// ---------------------------------------------------------------------------
__global__ __launch_bounds__(256) void qkv_gemm(const float* __restrict__ x,
                                                const float* __restrict__ w,
                                                float* __restrict__ qkv) {
    const int wave = (blockIdx.x * 256 + threadIdx.x) >> 5;
    const int lane = threadIdx.x & 31;
    const int n64  = wave & 63;
    const int rem  = wave >> 6;
    const int o32  = rem % 24;
    const int b    = rem / 24;

    const int m     = lane & 15;
    const int koff  = (lane >> 4) << 1;   // 0 or 2
    const int rbase = (lane >> 4) << 3;   // 0 or 8

    const float* W0 = w + (size_t)(o32 * 32 + m) * CCH;
    const float* W1 = W0 + 16 * CCH;
    const float* Xp = x + (size_t)b * CCH * NPIX + n64 * 64;

    v8f acc[2][4] = {};
    for (int c = 0; c < CCH; c += 4) {
        v2f a0, a1;
        a0.x = W0[c + koff]; a0.y = W0[c + koff + 1];
        a1.x = W1[c + koff]; a1.y = W1[c + koff + 1];
        const float* xr0 = Xp + (size_t)(c + koff) * NPIX;
        const float* xr1 = xr0 + NPIX;
#pragma unroll
        for (int j = 0; j < 4; ++j) {
            v2f bb;
            bb.x = xr0[j * 16 + m];
            bb.y = xr1[j * 16 + m];
            acc[0][j] = wmma4(a0, bb, acc[0][j]);
            acc[1][j] = wmma4(a1, bb, acc[1][j]);
        }
    }
    float* Out = qkv + (size_t)b * O3 * NPIX + n64 * 64;
#pragma unroll
    for (int i = 0; i < 2; ++i)
#pragma unroll
        for (int j = 0; j < 4; ++j)
#pragma unroll
            for (int v = 0; v < 8; ++v)
                Out[(size_t)(o32 * 32 + i * 16 + rbase + v) * NPIX + j * 16 + m] =
                    acc[i][j][v];
}

// ---------------------------------------------------------------------------
// Kernel 2a: q = softmax over d (DIM_HEAD) * QSCALE.  One thread per (b,h,n).
// ---------------------------------------------------------------------------
__global__ __launch_bounds__(256) void softmax_q(float* __restrict__ qkv) {
    const int id = blockIdx.x * 256 + threadIdx.x;   // 16*4*4096 threads
    const int n  = id & (NPIX - 1);
    const int bh = id >> 12;
    const int h  = bh & 3, b = bh >> 2;
    float* p = qkv + ((size_t)b * O3 + h * DIM_HEAD) * NPIX + n;

    float vals[DIM_HEAD];
    float mx = -INFINITY;
#pragma unroll
    for (int d = 0; d < DIM_HEAD; ++d) {
        vals[d] = p[(size_t)d * NPIX];
        mx = fmaxf(mx, vals[d]);
    }
    float s = 0.f;
#pragma unroll
    for (int d = 0; d < DIM_HEAD; ++d) {
        vals[d] = __expf(vals[d] - mx);
        s += vals[d];
    }
    const float inv = QSCALE / s;
#pragma unroll
    for (int d = 0; d < DIM_HEAD; ++d)
        p[(size_t)d * NPIX] = vals[d] * inv;
}

// ---------------------------------------------------------------------------
// Kernel 2b: k = softmax over n (4096).  One 256-thread block per (b,h,d) row.
// ---------------------------------------------------------------------------
__global__ __launch_bounds__(256) void softmax_k(float* __restrict__ qkv) {
    const int row = blockIdx.x;              // 16*4*64 = 4096 rows
    const int d = row & 63;
    const int h = (row >> 6) & 3;
    const int b = row >> 8;
    float* p = qkv + ((size_t)b * O3 + CCH + h * DIM_HEAD + d) * NPIX;

    __shared__ float red[256];
    const int t = threadIdx.x;
    float vals[16];
    float mx = -INFINITY;
#pragma unroll
    for (int i = 0; i < 16; ++i) {
        vals[i] = p[t + i * 256];
        mx = fmaxf(mx, vals[i]);
    }
    red[t] = mx;
    __syncthreads();
    for (int s = 128; s > 0; s >>= 1) {
        if (t < s) red[t] = fmaxf(red[t], red[t + s]);
        __syncthreads();
    }
    mx = red[0];
    __syncthreads();
    float sum = 0.f;
#pragma unroll
    for (int i = 0; i < 16; ++i) {
        vals[i] = __expf(vals[i] - mx);
        sum += vals[i];
    }
    red[t] = sum;
    __syncthreads();
    for (int s = 128; s > 0; s >>= 1) {
        if (t < s) red[t] += red[t + s];
        __syncthreads();
    }
    const float inv = 1.f / red[0];
#pragma unroll
    for (int i = 0; i < 16; ++i)
        p[t + i * 256] = vals[i] * inv;
}

// ---------------------------------------------------------------------------
// Kernel 3: ctx[b,h,d,e] = sum_n k[d,n] * v[e,n].  One block per (b,h).
// k/v chunks staged straight into LDS via GLOBAL_LOAD_ASYNC_TO_LDS_B128
// (ASYNCcnt), rows padded to 68 floats (16B-aligned, bank-conflict-free).
// 8 waves x 2 tiles = 16 WMMA tiles of 16x16, K = 4096.
// ---------------------------------------------------------------------------
__global__ __launch_bounds__(256) void kv_context(const float* __restrict__ qkv,
                                                  float* __restrict__ ctx) {
    const int bh = blockIdx.x;               // 64 blocks
    const int h = bh & 3, b = bh >> 2;
    const float* Kp = qkv + ((size_t)b * O3 + CCH + h * DIM_HEAD) * NPIX;
    const float* Vp = qkv + ((size_t)b * O3 + 2 * CCH + h * DIM_HEAD) * NPIX;

    __shared__ float lk[64 * PAD];
    __shared__ float lv[64 * PAD];

    const int t = threadIdx.x;
    const int wave = t >> 5, lane = t & 31;
    const int m = lane & 15;
    const int koff  = (lane >> 4) << 1;
    const int rbase = (lane >> 4) << 3;

    v8f acc[2] = {};
    for (int nb = 0; nb < NPIX; nb += 64) {
        __syncthreads();   // all waves done reading previous chunk
        // 64x64 floats per matrix = 1024 float4; 256 threads x 4 each
#pragma unroll
        for (int i = 0; i < 4; ++i) {
            const int f  = t + i * 256;       // float4 index
            const int r  = f >> 4;            // row (16 float4 per row)
            const int c4 = f & 15;
            const float* gk = Kp + (size_t)r * NPIX + nb + c4 * 4;
            const float* gv = Vp + (size_t)r * NPIX + nb + c4 * 4;
            const unsigned lko = lds_off(&lk[r * PAD + c4 * 4]);
            const unsigned lvo = lds_off(&lv[r * PAD + c4 * 4]);
            asm volatile("global_load_async_to_lds_b128 %0, %1, off"
                         :: "v"(lko), "v"(gk) : "memory");
            asm volatile("global_load_async_to_lds_b128 %0, %1, off"
                         :: "v"(lvo), "v"(gv) : "memory");
        }
        asm volatile("s_wait_asynccnt 0x0" ::: "memory");
        __syncthreads();   // everyone's async copies landed in LDS
#pragma unroll
        for (int tt = 0; tt < 2; ++tt) {
            const int tile = wave + tt * 8;       // 0..15
            const int di = tile >> 2, ej = tile & 3;
            v8f a8 = acc[tt];
            for (int kk = 0; kk < 64; kk += 4) {
                v2f a, bb;
                a.x  = lk[(di * 16 + m) * PAD + kk + koff];
                a.y  = lk[(di * 16 + m) * PAD + kk + koff + 1];
                bb.x = lv[(ej * 16 + m) * PAD + kk + koff];
                bb.y = lv[(ej * 16 + m) * PAD + kk + koff + 1];
                a8 = wmma4(a, bb, a8);
            }
            acc[tt] = a8;
        }
    }
    float* Cp = ctx + (size_t)bh * 64 * 64;      // [d][e]
#pragma unroll
    for (int tt = 0; tt < 2; ++tt) {
        const int tile = wave + tt * 8;
        const int di = tile >> 2, ej = tile & 3;
#pragma unroll
        for (int v = 0; v < 8; ++v)
            Cp[(di * 16 + rbase + v) * 64 + ej * 16 + m] = acc[tt][v];
    }
}

// ---------------------------------------------------------------------------
// Kernel 4: attn[b, h*64+e, n] = sum_d ctx[b,h,d,e] * q[b,h,d,n]
// one wave -> 32(e) x 64(n); waves = 16*4*2*64 = 8192; 1024 blocks
// ---------------------------------------------------------------------------
__global__ __launch_bounds__(256) void apply_ctx(const float* __restrict__ qkv,
                                                 const float* __restrict__ ctx,
                                                 float* __restrict__ attn) {
    const int wave = (blockIdx.x * 256 + threadIdx.x) >> 5;
    const int lane = threadIdx.x & 31;
    const int n64 = wave & 63;
    int rem = wave >> 6;
    const int e32 = rem & 1;  rem >>= 1;
    const int h = rem & 3;
    const int b = rem >> 2;

    const int m     = lane & 15;
    const int koff  = (lane >> 4) << 1;
    const int rbase = (lane >> 4) << 3;

    const float* Q  = qkv + ((size_t)b * O3 + h * DIM_HEAD) * NPIX + n64 * 64;
    const float* Cp = ctx + (size_t)(b * 4 + h) * 64 * 64;
    const int e0 = e32 * 32;

    v8f acc[2][4] = {};
    for (int d = 0; d < 64; d += 4) {
        v2f a0, a1;                              // A[e][d] = ctx[d][e]
        a0.x = Cp[(d + koff) * 64 + e0 + m];
        a0.y = Cp[(d + koff + 1) * 64 + e0 + m];
        a1.x = Cp[(d + koff) * 64 + e0 + 16 + m];
        a1.y = Cp[(d + koff + 1) * 64 + e0 + 16 + m];
        const float* q0 = Q + (size_t)(d + koff) * NPIX;
        const float* q1 = q0 + NPIX;
#pragma unroll
        for (int j = 0; j < 4; ++j) {
            v2f bb;
            bb.x = q0[j * 16 + m];
            bb.y = q1[j * 16 + m];
            acc[0][j] = wmma4(a0, bb, acc[0][j]);
            acc[1][j] = wmma4(a1, bb, acc[1][j]);
        }
    }
    float* Out = attn + ((size_t)b * CCH + h * DIM_HEAD) * NPIX + n64 * 64;
#pragma unroll
    for (int i = 0; i < 2; ++i)
#pragma unroll
        for (int j = 0; j < 4; ++j)
#pragma unroll
            for (int v = 0; v < 8; ++v)
                Out[(size_t)(e0 + i * 16 + rbase + v) * NPIX + j * 16 + m] =
                    acc[i][j][v];
}

// ---------------------------------------------------------------------------
// Kernel 5: y[b,o,n] = sum_c w_out[o,c] * attn[b,c,n] + b_out[o]
// one wave -> 32(o) x 64(n); waves = 16*8*64 = 8192; 1024 blocks
// ---------------------------------------------------------------------------
__global__ __launch_bounds__(256) void out_gemm(const float* __restrict__ attn,
                                                const float* __restrict__ w,
                                                const float* __restrict__ bias,
                                                float* __restrict__ y) {
    const int wave = (blockIdx.x * 256 + threadIdx.x) >> 5;
    const int lane = threadIdx.x & 31;
    const int n64 = wave & 63;
    const int rem = wave >> 6;
    const int o32 = rem & 7;
    const int b   = rem >> 3;

    const int m     = lane & 15;
    const int koff  = (lane >> 4) << 1;
    const int rbase = (lane >> 4) << 3;

    const float* W0 = w + (size_t)(o32 * 32 + m) * CCH;
    const float* W1 = W0 + 16 * CCH;
    const float* Ap = attn + (size_t)b * CCH * NPIX + n64 * 64;

    v8f acc[2][4] = {};
    for (int c = 0; c < CCH; c += 4) {
        v2f a0, a1;
        a0.x = W0[c + koff]; a0.y = W0[c + koff + 1];
        a1.x = W1[c + koff]; a1.y = W1[c + koff + 1];
        const float* ar0 = Ap + (size_t)(c + koff) * NPIX;
        const float* ar1 = ar0 + NPIX;
#pragma unroll
        for (int j = 0; j < 4; ++j) {
            v2f bb;
            bb.x = ar0[j * 16 + m];
            bb.y = ar1[j * 16 + m];
            acc[0][j] = wmma4(a0, bb, acc[0][j]);
            acc[1][j] = wmma4(a1, bb, acc[1][j]);
        }
    }
    float* Out = y + (size_t)b * CCH * NPIX + n64 * 64;
#pragma unroll
    for (int i = 0; i < 2; ++i)
#pragma unroll
        for (int j = 0; j < 4; ++j)
#pragma unroll
            for (int v = 0; v < 8; ++v) {
                const int o = o32 * 32 + i * 16 + rbase + v;
                Out[(size_t)o * NPIX + j * 16 + m] = acc[i][j][v] + bias[o];
            }
}

extern "C" void kernel_launch(void* const* d_in, const int* in_sizes, int n_in,
                              void* d_out, int out_size, void* d_ws, size_t ws_size,
                              hipStream_t stream) {
    const float* x     = (const float*)d_in[0];   // (16,256,64,64)
    const float* w_qkv = (const float*)d_in[1];   // (768,256)
    const float* w_out = (const float*)d_in[2];   // (256,256)
    const float* b_out = (const float*)d_in[3];   // (256,)
    float* out = (float*)d_out;                   // (16,256,64,64)

    float* qkv  = (float*)d_ws;                               // 192 MiB
    float* attn = qkv + (size_t)NB * O3 * NPIX;               //  64 MiB
    float* ctx  = attn + (size_t)NB * CCH * NPIX;             //   1 MiB

    qkv_gemm <<<3072, 256, 0, stream>>>(x, w_qkv, qkv);
    softmax_q<<<1024, 256, 0, stream>>>(qkv);
    softmax_k<<<4096, 256, 0, stream>>>(qkv);
    kv_context<<<64,  256, 0, stream>>>(qkv, ctx);
    apply_ctx<<<1024, 256, 0, stream>>>(qkv, ctx, attn);
    out_gemm <<<1024, 256, 0, stream>>>(attn, w_out, b_out, out);
}